// GraphAttention_6073083757198
// MI455X (gfx1250) — compile-verified
//
#include <hip/hip_runtime.h>
#include <cstdint>
#include <cstddef>

// Problem constants (from reference)
#define B_ 4
#define S_ 512
#define HID_ 256
#define NH_ 8
#define KS_ 32
#define KR_ 8
#define VS_ 32
#define RS_ 16
#define TOTAL_ 104            // 2*KS + VS + KR
#define NOUT_ (NH_ * TOTAL_)  // 832

typedef __attribute__((ext_vector_type(2))) float v2f;
typedef __attribute__((ext_vector_type(8))) float v8f;
typedef int v4i_ __attribute__((vector_size(16)));   // matches async-LDS builtin param

// ---- gfx1250 feature guards (also keep the host pass happy) ----
#if defined(__HIP_DEVICE_COMPILE__) && __has_builtin(__builtin_amdgcn_wmma_f32_16x16x4_f32)
#define HAVE_WMMA_F32 1
#else
#define HAVE_WMMA_F32 0
#endif

#if defined(__HIP_DEVICE_COMPILE__) && __has_builtin(__builtin_amdgcn_global_load_async_to_lds_b128)
#define HAVE_ASYNC_LDS 1
#else
#define HAVE_ASYNC_LDS 0
#endif

#define ASG __attribute__((address_space(1)))
#define ASL __attribute__((address_space(3)))

__device__ __forceinline__ float wave_max32(float v) {
  #pragma unroll
  for (int o = 16; o > 0; o >>= 1) v = fmaxf(v, __shfl_xor(v, o, 32));
  return v;
}
__device__ __forceinline__ float wave_sum32(float v) {
  #pragma unroll
  for (int o = 16; o > 0; o >>= 1) v += __shfl_xor(v, o, 32);
  return v;
}

// ============================================================================
// Kernel 1: H2 = H @ W_qkv^T  (2048 x 832, K=256), f32 WMMA 16x16x4.
// One wave per 16x16 tile. Scatter outputs into Qn/Kn/V/Qr per-head layouts.
// ============================================================================
__global__ __launch_bounds__(32) void k_qkv(const float* __restrict__ Hm,
                                            const float* __restrict__ Wqkv,
                                            float* __restrict__ Qn, float* __restrict__ Kn,
                                            float* __restrict__ Vv, float* __restrict__ Qr) {
  const int rt = blockIdx.x;       // row tile over B*S (128 tiles)
  const int ct = blockIdx.y;       // col tile over 832 (52 tiles)
  const int lane = threadIdx.x;
  const int mn = lane & 15;
  const int hi = lane >> 4;        // which K-half (WMMA f32 A/B layout)
  float vals[8];
#if HAVE_WMMA_F32
  v8f c = {0.f, 0.f, 0.f, 0.f, 0.f, 0.f, 0.f, 0.f};
  const float* ap = Hm   + (size_t)(rt * 16 + mn) * HID_ + 2 * hi;
  const float* bp = Wqkv + (size_t)(ct * 16 + mn) * HID_ + 2 * hi;
  for (int k0 = 0; k0 < HID_; k0 += 4) {
    v2f a, b;
    a.x = ap[k0]; a.y = ap[k0 + 1];     // A: M=mn, K = k0 + 2*hi + {0,1}
    b.x = bp[k0]; b.y = bp[k0 + 1];     // B: N=mn, K = k0 + 2*hi + {0,1} (B = Wqkv^T)
    c = __builtin_amdgcn_wmma_f32_16x16x4_f32(false, a, false, b, (short)0, c, false, false);
  }
  #pragma unroll
  for (int r = 0; r < 8; ++r) vals[r] = c[r];
#else
  float acc[8] = {0, 0, 0, 0, 0, 0, 0, 0};
  for (int k = 0; k < HID_; ++k) {
    const float bw = Wqkv[(size_t)(ct * 16 + mn) * HID_ + k];
    #pragma unroll
    for (int r = 0; r < 8; ++r)
      acc[r] += Hm[(size_t)(rt * 16 + r + 8 * hi) * HID_ + k] * bw;
  }
  #pragma unroll
  for (int r = 0; r < 8; ++r) vals[r] = acc[r];
#endif
  const int o = ct * 16 + mn;        // output channel in [0, 832)
  const int h = o / TOTAL_;
  const int t = o - h * TOTAL_;
  #pragma unroll
  for (int r = 0; r < 8; ++r) {
    const int row = rt * 16 + r + 8 * hi;   // C/D layout: lanes 16-31 hold M=r+8
    const int b = row / S_;
    const int s = row - b * S_;
    const size_t base = (size_t)(b * NH_ + h) * S_ + s;
    const float val = vals[r];
    if (t < KS_)                  Qn[base * KS_ + t] = val;
    else if (t < 2 * KS_)         Kn[base * KS_ + (t - KS_)] = val;
    else if (t < 2 * KS_ + VS_)   Vv[base * VS_ + (t - 2 * KS_)] = val;
    else                          Qr[base * KR_ + (t - 2 * KS_ - VS_)] = val;
  }
}

// ============================================================================
// Kernel 2: qr_eff[b,h,s,:] = Wd_h^T @ Qr[b,h,s,:]   (RS outputs, KR=8 MACs)
// ============================================================================
__global__ __launch_bounds__(256) void k_qreff(const float* __restrict__ Qr,
                                               const float* __restrict__ Wd,
                                               float* __restrict__ Qe) {
  const int idx = blockIdx.x * 256 + threadIdx.x;   // over B*NH*S*RS
  const int r = idx & (RS_ - 1);
  const int row = idx >> 4;                         // (b*NH+h)*S + s
  const int h = (row / S_) % NH_;
  const float* q = Qr + (size_t)row * KR_;
  float acc = 0.f;
  #pragma unroll
  for (int k = 0; k < KR_; ++k) acc += q[k] * Wd[(h * KR_ + k) * RS_ + r];
  Qe[idx] = acc;
}

// ============================================================================
// Kernel 3: attention. One block per (b,i); 8 waves = 8 heads.
// route[b,i,:,:] (32 KB) staged once into LDS via async-to-LDS, shared by all
// heads.  scores -> softmax (wave32 shuffle reduce) -> p@V + (Wroute@(p@route)).
// ============================================================================
__global__ __launch_bounds__(256) void k_attn(const float* __restrict__ Qn,
                                              const float* __restrict__ Kn,
                                              const float* __restrict__ Vv,
                                              const float* __restrict__ Qe,
                                              const float* __restrict__ route,
                                              const float* __restrict__ mask,
                                              const float* __restrict__ Wroute,
                                              float* __restrict__ Ctx) {
  __shared__ float lds_route[S_ * RS_];   // 32 KB: route row for this (b,i)
  __shared__ float lds_probs[NH_][S_];    // 16 KB
  __shared__ float lds_q[NH_][KS_];
  __shared__ float lds_qe[NH_][RS_];
  __shared__ float lds_racc[NH_][RS_];

  const int bi = blockIdx.x;            // 0..B*S-1
  const int b = bi / S_;
  const int i = bi - b * S_;
  const int tid = threadIdx.x;
  const int lane = tid & 31;
  const int h = tid >> 5;               // wave id == head

  // --- stage route[b,i,:,:] into LDS (2048 x 16B chunks, 8 per thread) ---
  const float* rsrc = route + (size_t)bi * S_ * RS_;
#if HAVE_ASYNC_LDS
  #pragma unroll
  for (int cch = 0; cch < 8; ++cch) {
    const int off = (cch * 256 + tid) * 4;   // float4 granularity
    __builtin_amdgcn_global_load_async_to_lds_b128(
        (ASG v4i_*)(rsrc + off), (ASL v4i_*)(lds_route + off), 0, 0);
  }
#if defined(__HIP_DEVICE_COMPILE__) && __has_builtin(__builtin_amdgcn_s_wait_asynccnt)
  __builtin_amdgcn_s_wait_asynccnt(0);
#else
  asm volatile("s_wait_asynccnt 0" ::: "memory");
#endif
#else
  #pragma unroll
  for (int cch = 0; cch < 8; ++cch) {
    const int off = (cch * 256 + tid) * 4;
    *reinterpret_cast<float4*>(lds_route + off) =
        *reinterpret_cast<const float4*>(rsrc + off);
  }
#endif
  if (lane < KS_) lds_q[h][lane]  = Qn[((size_t)(b * NH_ + h) * S_ + i) * KS_ + lane];
  if (lane < RS_) lds_qe[h][lane] = Qe[((size_t)(b * NH_ + h) * S_ + i) * RS_ + lane];
  __syncthreads();

  // --- scores: lane handles j = jt*32 + lane ---
  const float scl = rsqrtf((float)(KS_ + KR_));
  const float* KnH = Kn + (size_t)(b * NH_ + h) * S_ * KS_;
  float sc[16];
  float mx = -3.4e38f;
  #pragma unroll 4
  for (int jt = 0; jt < 16; ++jt) {
    const int j = jt * 32 + lane;
    const float* kp = KnH + (size_t)j * KS_;
    float s = 0.f;
    #pragma unroll
    for (int k = 0; k < KS_; ++k) s += lds_q[h][k] * kp[k];
    const float* rp = lds_route + j * RS_;
    #pragma unroll
    for (int r = 0; r < RS_; ++r) s += lds_qe[h][r] * rp[r];
    s = s * scl + mask[b * S_ + j];
    sc[jt] = s;
    mx = fmaxf(mx, s);
  }
  mx = wave_max32(mx);
  float sum = 0.f;
  #pragma unroll
  for (int jt = 0; jt < 16; ++jt) { const float p = __expf(sc[jt] - mx); sc[jt] = p; sum += p; }
  sum = wave_sum32(sum);
  const float inv = 1.f / sum;
  #pragma unroll
  for (int jt = 0; jt < 16; ++jt) lds_probs[h][jt * 32 + lane] = sc[jt] * inv;
  __syncthreads();

  // --- ctx[v] = sum_j p_j * V[j,v] (lane owns v = lane) ---
  const float* VH = Vv + (size_t)(b * NH_ + h) * S_ * VS_;
  float cacc = 0.f;
  for (int j = 0; j < S_; ++j) cacc += lds_probs[h][j] * VH[(size_t)j * VS_ + lane];

  // --- r_acc[r] = sum_j p_j * route[j,r] (lanes 0..15) ---
  if (lane < RS_) {
    float racc = 0.f;
    for (int j = 0; j < S_; ++j) racc += lds_probs[h][j] * lds_route[j * RS_ + lane];
    lds_racc[h][lane] = racc;
  }
  __syncthreads();

  // --- ctx += Wroute_h @ r_acc ---
  float c2 = 0.f;
  #pragma unroll
  for (int r = 0; r < RS_; ++r) c2 += Wroute[(h * VS_ + lane) * RS_ + r] * lds_racc[h][r];

  Ctx[(size_t)bi * (NH_ * VS_) + h * VS_ + lane] = cacc + c2;
}

// ============================================================================
// Kernel 4: OutPre = Ctx @ W_out^T + b_out   (2048 x 256, K=256), f32 WMMA.
// ============================================================================
__global__ __launch_bounds__(32) void k_outproj(const float* __restrict__ Ctx,
                                                const float* __restrict__ Wout,
                                                const float* __restrict__ bout,
                                                float* __restrict__ Op) {
  const int rt = blockIdx.x;     // 128 row tiles
  const int ct = blockIdx.y;     // 16 col tiles
  const int lane = threadIdx.x;
  const int mn = lane & 15;
  const int hi = lane >> 4;
  float vals[8];
#if HAVE_WMMA_F32
  v8f c = {0.f, 0.f, 0.f, 0.f, 0.f, 0.f, 0.f, 0.f};
  const float* ap = Ctx  + (size_t)(rt * 16 + mn) * (NH_ * VS_) + 2 * hi;
  const float* bp = Wout + (size_t)(ct * 16 + mn) * (NH_ * VS_) + 2 * hi;
  for (int k0 = 0; k0 < NH_ * VS_; k0 += 4) {
    v2f a, b;
    a.x = ap[k0]; a.y = ap[k0 + 1];
    b.x = bp[k0]; b.y = bp[k0 + 1];
    c = __builtin_amdgcn_wmma_f32_16x16x4_f32(false, a, false, b, (short)0, c, false, false);
  }
  #pragma unroll
  for (int r = 0; r < 8; ++r) vals[r] = c[r];
#else
  float acc[8] = {0, 0, 0, 0, 0, 0, 0, 0};
  for (int k = 0; k < NH_ * VS_; ++k) {
    const float bw = Wout[(size_t)(ct * 16 + mn) * (NH_ * VS_) + k];
    #pragma unroll
    for (int r = 0; r < 8; ++r)
      acc[r] += Ctx[(size_t)(rt * 16 + r + 8 * hi) * (NH_ * VS_) + k] * bw;
  }
  #pragma unroll
  for (int r = 0; r < 8; ++r) vals[r] = acc[r];
#endif
  const int n = ct * 16 + mn;
  const float bb = bout[n];
  #pragma unroll
  for (int r = 0; r < 8; ++r)
    Op[(size_t)(rt * 16 + r + 8 * hi) * HID_ + n] = vals[r] + bb;
}

// ============================================================================
// Kernel 5: LayerNorm + residual. One block (256 threads) per row.
// ============================================================================
__global__ __launch_bounds__(256) void k_ln(const float* __restrict__ Op,
                                            const float* __restrict__ Hm,
                                            const float* __restrict__ lnw,
                                            const float* __restrict__ lnb,
                                            float* __restrict__ out) {
  __shared__ float red[HID_];
  const int row = blockIdx.x;
  const int c = threadIdx.x;
  const float v = Op[(size_t)row * HID_ + c];
  red[c] = v;
  __syncthreads();
  for (int s = HID_ / 2; s > 0; s >>= 1) {
    if (c < s) red[c] += red[c + s];
    __syncthreads();
  }
  const float u = red[0] * (1.f / HID_);
  __syncthreads();
  const float d = v - u;
  red[c] = d * d;
  __syncthreads();
  for (int s = HID_ / 2; s > 0; s >>= 1) {
    if (c < s) red[c] += red[c + s];
    __syncthreads();
  }
  const float var = red[0] * (1.f / HID_);
  const float ln = lnw[c] * d * rsqrtf(var + 1e-12f) + lnb[c];
  out[(size_t)row * HID_ + c] = ln + Hm[(size_t)row * HID_ + c];
}

// ============================================================================
// Launch.  Workspace layout (floats): Qn | Kn | V | Qr | QrEff | Ctx | OutPre
// Total 3,014,656 floats = ~11.5 MB.
// ============================================================================
extern "C" void kernel_launch(void* const* d_in, const int* in_sizes, int n_in,
                              void* d_out, int out_size, void* d_ws, size_t ws_size,
                              hipStream_t stream) {
  const float* Hm     = (const float*)d_in[0];
  const float* route  = (const float*)d_in[1];
  const float* mask   = (const float*)d_in[2];
  const float* Wqkv   = (const float*)d_in[3];
  const float* Wd     = (const float*)d_in[4];
  const float* Wroute = (const float*)d_in[5];
  const float* Wout   = (const float*)d_in[6];
  const float* bout   = (const float*)d_in[7];
  const float* lnw    = (const float*)d_in[8];
  const float* lnb    = (const float*)d_in[9];
  float* out = (float*)d_out;

  float* ws  = (float*)d_ws;
  float* Qn  = ws;
  float* Kn  = Qn  + (size_t)B_ * NH_ * S_ * KS_;
  float* Vv  = Kn  + (size_t)B_ * NH_ * S_ * KS_;
  float* Qr  = Vv  + (size_t)B_ * NH_ * S_ * VS_;
  float* Qe  = Qr  + (size_t)B_ * NH_ * S_ * KR_;
  float* Ctx = Qe  + (size_t)B_ * NH_ * S_ * RS_;
  float* Op  = Ctx + (size_t)B_ * S_ * NH_ * VS_;

  k_qkv<<<dim3((B_ * S_) / 16, NOUT_ / 16), 32, 0, stream>>>(Hm, Wqkv, Qn, Kn, Vv, Qr);
  k_qreff<<<(B_ * NH_ * S_ * RS_) / 256, 256, 0, stream>>>(Qr, Wd, Qe);
  k_attn<<<B_ * S_, 256, 0, stream>>>(Qn, Kn, Vv, Qe, route, mask, Wroute, Ctx);
  k_outproj<<<dim3((B_ * S_) / 16, HID_ / 16), 32, 0, stream>>>(Ctx, Wout, bout, Op);
  k_ln<<<B_ * S_, HID_, 0, stream>>>(Op, Hm, lnw, lnb, out);
}